// YOLOLayer_9070970929599
// MI455X (gfx1250) — compile-verified
//
#include <hip/hip_runtime.h>
#include <hip/hip_bf16.h>
#include <math.h>

// ---------------------------------------------------------------------------
// YOLO layer loss for MI455X (gfx1250, wave32).
// Pipeline:
//   K1 yolo_setup  : 32 blocks x 32 thr  - per-GT anchor match + target calc
//   K2 yolo_main   : 1536 blocks x 256   - per-cell conf/iou/loss, block reduce
//   K3 yolo_reduce : 1 block x 32 (wave) - WMMA-based exact f32 reduction
// Only ~8MB of input is read (cls touched only at GT cells) -> HBM-bound.
// Deterministic: fixed iteration order, no atomics.
// ---------------------------------------------------------------------------

#define NB   32
#define NA   3
#define NH   64
#define NW   64
#define NG   24
#define NCLS 80
#define STRIDE_F 8.0f
#define IGNORE_THRE 0.6f

#define CELLS_PER_IMG (NA * NH * NW)          // 12288
#define BLOCKS_PER_IMG (CELLS_PER_IMG / 256)  // 48
#define NBLOCKS (NB * BLOCKS_PER_IMG)         // 1536

// GT table entry layout (16 floats):
// 0:cx 1:cy 2:w 3:h 4:valid 5:tn 6:tj 7:ti 8:cid 9..12:upd 13:weight 14,15:pad
#define GT_STRIDE 16

typedef float v2f __attribute__((ext_vector_type(2)));
typedef float v8f __attribute__((ext_vector_type(8)));

__device__ __forceinline__ float bce(float x, float t) {
    return fmaxf(x, 0.0f) - x * t + log1pf(expf(-fabsf(x)));
}

__device__ __forceinline__ float iou_cwh1(float px, float py, float pw, float ph,
                                          float cx, float cy, float gw, float gh) {
    float x1 = fmaxf(px - pw * 0.5f, cx - gw * 0.5f);
    float y1 = fmaxf(py - ph * 0.5f, cy - gh * 0.5f);
    float x2 = fminf(px + pw * 0.5f, cx + gw * 0.5f);
    float y2 = fminf(py + ph * 0.5f, cy + gh * 0.5f);
    float iw = fmaxf(x2 - x1, 0.0f);
    float ih = fmaxf(y2 - y1, 0.0f);
    float inter = iw * ih;
    return inter / (pw * ph + gw * gh - inter);
}

// -------------------------- K1: per-GT setup -------------------------------
__global__ void yolo_setup(const float* __restrict__ labels,
                           const int* __restrict__ img_h,
                           const int* __restrict__ img_w,
                           float* __restrict__ gt_table,
                           float* __restrict__ anyv) {
    const float AW[9] = {10.f, 16.f, 33.f, 30.f, 62.f, 59.f, 116.f, 156.f, 373.f};
    const float AH[9] = {13.f, 30.f, 23.f, 61.f, 45.f, 119.f, 90.f, 198.f, 326.f};
    int b = blockIdx.x;
    int g = threadIdx.x;
    bool v = false;
    if (g < NG) {
        const float* lab = labels + (size_t)(b * NG + g) * 5;
        float cid = lab[0];
        float cx = lab[1], cy = lab[2];
        float w = lab[3], h = lab[4];
        // argmax over 9 anchor IoUs (centered boxes => closed form), first-max wins
        int best = 0;
        float bestv = -1.0f;
        float area_g = w * h;
        #pragma unroll
        for (int i = 0; i < 9; ++i) {
            float inter = fminf(w, AW[i]) * fminf(h, AH[i]);
            float iou = inter / (area_g + AW[i] * AH[i] - inter);
            if (iou > bestv) { bestv = iou; best = i; }
        }
        bool valid = (best < NA);           // ANCH_LO == 0
        int tn = best % NA;
        float gx = cx / STRIDE_F, gy = cy / STRIDE_F;
        int ti = (int)gx, tj = (int)gy;
        float aw = AW[tn], ah = AH[tn];
        float img_area = (float)(*img_h) * (float)(*img_w);
        float* e = gt_table + (size_t)(b * NG + g) * GT_STRIDE;
        e[0] = cx; e[1] = cy; e[2] = w; e[3] = h;
        e[4] = valid ? 1.0f : 0.0f;
        e[5] = (float)tn; e[6] = (float)tj; e[7] = (float)ti;
        e[8] = cid;
        e[9]  = gx - floorf(gx);
        e[10] = gy - floorf(gy);
        e[11] = logf(w / aw + 1e-8f);
        e[12] = logf(h / ah + 1e-8f);
        e[13] = 2.0f - area_g / img_area;
        e[14] = 0.0f; e[15] = 0.0f;
        v = valid;
    }
    bool any = __any((int)v);
    if (g == 0) anyv[b] = any ? 1.0f : 0.0f;
}

// -------------------------- K2: per-cell pass ------------------------------
__global__ void yolo_main(const float* __restrict__ bbox,
                          const float* __restrict__ conf,
                          const float* __restrict__ cls,
                          const float* __restrict__ gt_table,
                          const float* __restrict__ anyv,
                          float* __restrict__ partials) {
    const float AWA[NA] = {10.f, 16.f, 33.f};
    const float AHA[NA] = {13.f, 30.f, 23.f};

    __shared__ float sgt[NG * GT_STRIDE];   // 24 x 16 floats
    __shared__ float s_anyv;
    __shared__ float sred[256];

    int tid = threadIdx.x;
    int b = blockIdx.x / BLOCKS_PER_IMG;
    int local = (blockIdx.x % BLOCKS_PER_IMG) * 256 + tid;   // [0, 12288)
    int a = local / (NH * NW);
    int rem = local % (NH * NW);
    int y = rem / NW;
    int x = rem % NW;

    // stage this image's GT table into LDS
    const float* gsrc = gt_table + (size_t)b * NG * GT_STRIDE;
    for (int i = tid; i < NG * GT_STRIDE; i += 256) sgt[i] = gsrc[i];
    if (tid == 0) s_anyv = anyv[b];
    __syncthreads();

    int cell = ((b * NA + a) * NH + y) * NW + x;

    // predicted box
    const float4 t4 = ((const float4*)bbox)[cell];
    float tx = t4.x, ty = t4.y, tw = t4.z, th = t4.w;
    float px = (1.0f / (1.0f + expf(-tx)) + (float)x) * STRIDE_F;
    float py = (1.0f / (1.0f + expf(-ty)) + (float)y) * STRIDE_F;
    float pw = expf(tw) * AWA[a];
    float ph = expf(th) * AHA[a];

    // max IoU vs all 24 GT boxes + GT-cell matching (last match wins; cid set)
    float iou_max = 0.0f;
    int last_match = -1;
    unsigned cm0 = 0u, cm1 = 0u, cm2 = 0u;
    #pragma unroll 4
    for (int g = 0; g < NG; ++g) {
        const float* e = &sgt[g * GT_STRIDE];
        iou_max = fmaxf(iou_max, iou_cwh1(px, py, pw, ph, e[0], e[1], e[2], e[3]));
        bool m = (e[4] > 0.5f) && ((int)e[5] == a) && ((int)e[6] == y) && ((int)e[7] == x);
        if (m) {
            last_match = g;
            int cid = (int)e[8];
            if (cid < 32)      cm0 |= 1u << cid;
            else if (cid < 64) cm1 |= 1u << (cid - 32);
            else               cm2 |= 1u << (cid - 64);
        }
    }
    bool is_gt = (last_match >= 0);
    bool cmask = (s_anyv > 0.5f) ? (iou_max < IGNORE_THRE) : true;
    if (is_gt) cmask = true;

    float tot = 0.0f;
    float cv = conf[cell];
    if (cmask) tot += bce(cv, is_gt ? 1.0f : 0.0f);

    if (is_gt) {
        const float* e = &sgt[last_match * GT_STRIDE];
        float wgt = e[13];
        tot += wgt * (bce(tx, e[9]) + bce(ty, e[10]));
        float dw = tw - e[11], dh = th - e[12];
        tot += 0.5f * wgt * (dw * dw + dh * dh);
        const float* cp = cls + (size_t)cell * NCLS;
        for (int c = 0; c < 32; ++c)  tot += bce(cp[c],      ((cm0 >> c) & 1u) ? 1.0f : 0.0f);
        for (int c = 0; c < 32; ++c)  tot += bce(cp[c + 32], ((cm1 >> c) & 1u) ? 1.0f : 0.0f);
        for (int c = 0; c < 16; ++c)  tot += bce(cp[c + 64], ((cm2 >> c) & 1u) ? 1.0f : 0.0f);
    }

    // deterministic block reduction
    sred[tid] = tot;
    __syncthreads();
    for (int s = 128; s > 0; s >>= 1) {
        if (tid < s) sred[tid] += sred[tid + s];
        __syncthreads();
    }
    if (tid == 0) partials[blockIdx.x] = sred[0];
}

// ---------------- K3: single-wave final reduction via WMMA -----------------
// 1536 partials -> 32 lane accumulators -> V_WMMA_F32_16X16X4_F32 with an
// all-ones B matrix: D[m][n] = s_m + s_{m+16} (broadcast over n). Summing the
// 8 C/D VGPRs gives each half-wave's subtotal; shfl_xor(16) combines halves.
__global__ void yolo_reduce(const float* __restrict__ partials,
                            float* __restrict__ out) {
    int lane = threadIdx.x;   // 32 threads, EXEC all ones (WMMA requirement)
    float s = 0.0f;
    #pragma unroll 4
    for (int k = 0; k < NBLOCKS / 32; ++k)
        s += partials[lane + 32 * k];

    v2f amat; amat.x = s;    amat.y = 0.0f;   // A[m][0]=s_m, A[m][2]=s_{m+16}
    v2f bmat; bmat.x = 1.0f; bmat.y = 1.0f;   // B = all ones
    v8f c = {};
    c = __builtin_amdgcn_wmma_f32_16x16x4_f32(
            /*neg_a=*/false, amat, /*neg_b=*/false, bmat,
            /*c_mod=*/(short)0, c, /*reuse_a=*/false, /*reuse_b=*/false);

    float half_sum = c[0] + c[1] + c[2] + c[3] + c[4] + c[5] + c[6] + c[7];
    float total = half_sum + __shfl_xor(half_sum, 16, 32);
    if (lane == 0) out[0] = total;
}

// ---------------------------------------------------------------------------
extern "C" void kernel_launch(void* const* d_in, const int* in_sizes, int n_in,
                              void* d_out, int out_size, void* d_ws, size_t ws_size,
                              hipStream_t stream) {
    const float* bbox   = (const float*)d_in[0];
    const float* conf   = (const float*)d_in[1];
    const float* cls    = (const float*)d_in[2];
    const float* labels = (const float*)d_in[3];
    const int*   img_h  = (const int*)d_in[4];
    const int*   img_w  = (const int*)d_in[5];

    float* ws = (float*)d_ws;
    float* partials = ws;                                // 1536 floats
    float* gt_table = ws + 2048;                         // 32*24*16 = 12288 floats
    float* anyv     = ws + 2048 + NB * NG * GT_STRIDE;   // 32 floats

    yolo_setup<<<NB, 32, 0, stream>>>(labels, img_h, img_w, gt_table, anyv);
    yolo_main<<<NBLOCKS, 256, 0, stream>>>(bbox, conf, cls, gt_table, anyv, partials);
    yolo_reduce<<<1, 32, 0, stream>>>(partials, (float*)d_out);
}